// OnlineClustering_48163763258100
// MI455X (gfx1250) — compile-verified
//
#include <hip/hip_runtime.h>
#include <math.h>

typedef __attribute__((ext_vector_type(2))) float v2f;
typedef __attribute__((ext_vector_type(8))) float v8f;

#define S_DIM 4
#define B_DIM 4096
#define D_DIM 768
#define K_DIM 8192
#define M_TOT (S_DIM * B_DIM)          // 16384 rows total

#define INV_TT 20.0f                   // 1 / 0.05 (target temp)
#define INV_TP (1.0f / 0.12f)          // 1 / 0.12 (pred temp)
#define EXP_MAX_C 50.0f
#define SK_EPS_C 1e-8f
#define NORM_EPS_C 1e-7f
#define NEG_BIG (-3.402823466e+38f)

#define NCHUNK 32                      // b-chunks for column reductions
#define CHUNK (B_DIM / NCHUNK)         // 128 rows per chunk

// ---------------- wave32 / block reductions ----------------
__device__ __forceinline__ float waveSum(float v) {
#pragma unroll
  for (int o = 16; o > 0; o >>= 1) v += __shfl_xor(v, o, 32);
  return v;
}
__device__ __forceinline__ float waveMax(float v) {
#pragma unroll
  for (int o = 16; o > 0; o >>= 1) v = fmaxf(v, __shfl_xor(v, o, 32));
  return v;
}

// 256-thread block reductions (8 wave32 waves). Full-block participation only.
__device__ __forceinline__ float blockSum256(float v) {
  __shared__ float sm[9];
  const int lane = threadIdx.x & 31, w = threadIdx.x >> 5;
  v = waveSum(v);
  if (lane == 0) sm[w] = v;
  __syncthreads();
  if (w == 0) {
    float x = (lane < 8) ? sm[lane] : 0.0f;
    x = waveSum(x);
    if (lane == 0) sm[8] = x;
  }
  __syncthreads();
  float r = sm[8];
  __syncthreads();
  return r;
}
__device__ __forceinline__ float blockMax256(float v) {
  __shared__ float sm[9];
  const int lane = threadIdx.x & 31, w = threadIdx.x >> 5;
  v = waveMax(v);
  if (lane == 0) sm[w] = v;
  __syncthreads();
  if (w == 0) {
    float x = (lane < 8) ? sm[lane] : NEG_BIG;
    x = waveMax(x);
    if (lane == 0) sm[8] = x;
  }
  __syncthreads();
  float r = sm[8];
  __syncthreads();
  return r;
}

// ---------------- 1) per-row inverse L2 norm of x ----------------
__global__ __launch_bounds__(256) void row_invnorm_kernel(
    const float* __restrict__ x, float* __restrict__ invn) {
  const int m = blockIdx.x;
  const float* row = x + (size_t)m * D_DIM;
  float ss = 0.0f;
  for (int k = threadIdx.x; k < D_DIM; k += 256) {
    float v = row[k];
    ss += v * v;
  }
  ss = blockSum256(ss);
  if (threadIdx.x == 0) invn[m] = 1.0f / fmaxf(sqrtf(ss), NORM_EPS_C);
}

// ---------------- 2) WMMA f32 GEMM: logits = (x * invn) . W^T ----------------
// C[m][n] = sum_d x[m][d]*invn[m] * W[n][d];  M=16384, N=8192, K=768
#define BM 128
#define BN 128
#define BK 32
#define LDT 36   // padded LDS row (36*k mod 64 hits distinct banks)

__global__ __launch_bounds__(256) void gemm_wmma_f32(
    const float* __restrict__ X, const float* __restrict__ W,
    const float* __restrict__ invn, float* __restrict__ out) {
  __shared__ float As[BM][LDT];
  __shared__ float Bs[BN][LDT];

  const int tid = threadIdx.x;
  const int lane = tid & 31;
  const int wave = tid >> 5;
  const int hi = lane >> 4;       // half-wave select (K pairs / M+8 rows)
  const int l15 = lane & 15;
  const int m0 = blockIdx.y * BM;
  const int n0 = blockIdx.x * BN;
  const int mW = (wave & 3) * 32; // 4 waves stacked in M (32 rows each)
  const int nW = (wave >> 2) * 64;// 2 waves across N (64 cols each)

  v8f acc[2][4] = {};

  for (int kt = 0; kt < D_DIM; kt += BK) {
    // cooperative stage: 256 threads x 4 float4 each per tile
#pragma unroll
    for (int i = 0; i < 4; i++) {
      const int p = tid + 256 * i;   // 0..1023 float4 slots
      const int row = p >> 3;        // 8 float4 per 32-float row
      const int c4 = (p & 7) << 2;
      const float4 av = *(const float4*)(X + (size_t)(m0 + row) * D_DIM + kt + c4);
      const float s = invn[m0 + row];
      float4 an;
      an.x = av.x * s; an.y = av.y * s; an.z = av.z * s; an.w = av.w * s;
      *(float4*)&As[row][c4] = an;
      const float4 bv = *(const float4*)(W + (size_t)(n0 + row) * D_DIM + kt + c4);
      *(float4*)&Bs[row][c4] = bv;
    }
    if (kt + BK < D_DIM) {
      // hint next K-tile into cache (global_prefetch_b8)
      __builtin_prefetch(X + (size_t)(m0 + (tid >> 1)) * D_DIM + kt + BK + (tid & 1) * 16, 0, 0);
      __builtin_prefetch(W + (size_t)(n0 + (tid >> 1)) * D_DIM + kt + BK + (tid & 1) * 16, 0, 0);
    }
    __syncthreads();

#pragma unroll
    for (int kk = 0; kk < BK; kk += 4) {
      const int kA = kk + 2 * hi;  // VGPR0/1 hold K={0,1} (lanes 0-15), K={2,3} (lanes 16-31)
      v2f a[2], b[4];
#pragma unroll
      for (int mi = 0; mi < 2; mi++)
        a[mi] = *(const v2f*)&As[mW + mi * 16 + l15][kA];
#pragma unroll
      for (int ni = 0; ni < 4; ni++)
        b[ni] = *(const v2f*)&Bs[nW + ni * 16 + l15][kA];
#pragma unroll
      for (int mi = 0; mi < 2; mi++)
#pragma unroll
        for (int ni = 0; ni < 4; ni++)
          acc[mi][ni] = __builtin_amdgcn_wmma_f32_16x16x4_f32(
              false, a[mi], false, b[ni], (short)0, acc[mi][ni], false, false);
    }
    __syncthreads();
  }

  // epilogue: C layout -> VGPR i holds rows (i, i+8) for lane halves
#pragma unroll
  for (int mi = 0; mi < 2; mi++)
#pragma unroll
    for (int ni = 0; ni < 4; ni++) {
      const int rbase = m0 + mW + mi * 16 + hi * 8;
      const int col = n0 + nW + ni * 16 + l15;
#pragma unroll
      for (int i = 0; i < 8; i++)
        out[(size_t)(rbase + i) * K_DIM + col] = acc[mi][ni][i];
    }
}

// ---------------- 3) Sinkhorn column max (shift), per s-slice ----------------
__global__ __launch_bounds__(256) void colmax_part_kernel(
    const float* __restrict__ L, float* __restrict__ part) {
  const int k = blockIdx.x * 256 + threadIdx.x;
  const int b0 = blockIdx.y * CHUNK;
  const float* p = L + (size_t)b0 * K_DIM + k;
  float m = NEG_BIG;
  for (int b = 0; b < CHUNK; b++) m = fmaxf(m, p[(size_t)b * K_DIM]);
  part[(size_t)blockIdx.y * K_DIM + k] = m;
}

__global__ __launch_bounds__(256) void colmax_fin_kernel(
    const float* __restrict__ part, float* __restrict__ shift,
    float* __restrict__ cvec) {
  const int k = blockIdx.x * 256 + threadIdx.x;
  float m = NEG_BIG;
  for (int i = 0; i < NCHUNK; i++) m = fmaxf(m, part[(size_t)i * K_DIM + k]);
  shift[k] = m;
  cvec[k] = 1.0f;
}

__global__ __launch_bounds__(256) void init_r_kernel(float* __restrict__ rvec) {
  rvec[blockIdx.x * 256 + threadIdx.x] = 1.0f;
}

// colsum pass: S_k = sum_b r[b] * exp(L/tt + 50 - shift_k)
__global__ __launch_bounds__(256) void colsum_part_kernel(
    const float* __restrict__ L, const float* __restrict__ rvec,
    const float* __restrict__ shift, float* __restrict__ part) {
  const int k = blockIdx.x * 256 + threadIdx.x;
  const int b0 = blockIdx.y * CHUNK;
  const float* p = L + (size_t)b0 * K_DIM + k;
  const float sh = EXP_MAX_C - shift[k];
  float s = 0.0f;
  for (int b = 0; b < CHUNK; b++)
    s += rvec[b0 + b] * __expf(p[(size_t)b * K_DIM] * INV_TT + sh);
  part[(size_t)blockIdx.y * K_DIM + k] = s;
}

__global__ __launch_bounds__(256) void colsum_fin_kernel(
    const float* __restrict__ part, float* __restrict__ cvec) {
  const int k = blockIdx.x * 256 + threadIdx.x;
  float s = 0.0f;
  for (int i = 0; i < NCHUNK; i++) s += part[(size_t)i * K_DIM + k];
  const float cv = cvec[k];
  cvec[k] = cv / (cv * s + SK_EPS_C);   // col-normalize step folded into c
}

// rowsum pass: T_b = sum_k exp(...)*c[k]; update r[b]
__global__ __launch_bounds__(256) void rowsum_update_kernel(
    const float* __restrict__ L, const float* __restrict__ shift,
    const float* __restrict__ cvec, float* __restrict__ rvec) {
  const int b = blockIdx.x;
  const float* row = L + (size_t)b * K_DIM;
  float s = 0.0f;
#pragma unroll 4
  for (int j = 0; j < K_DIM / 256; j++) {
    const int k = threadIdx.x + 256 * j;
    s += __expf(row[k] * INV_TT + EXP_MAX_C - shift[k]) * cvec[k];
  }
  s = blockSum256(s);
  if (threadIdx.x == 0) {
    const float rb = rvec[b];
    rvec[b] = rb / (rb * s + SK_EPS_C);
  }
}

// ---------------- 4) final fused: assignments (in place) + per-row loss ----------------
__global__ __launch_bounds__(256) void final_fused_kernel(
    float* __restrict__ L, const float* __restrict__ shift,
    const float* __restrict__ cvec, const float* __restrict__ rvec,
    float* __restrict__ lossRows, int rowOffset) {
  const int b = blockIdx.x;
  float* row = L + (size_t)b * K_DIM;

  float Lv[K_DIM / 256];
  float mx = NEG_BIG;
#pragma unroll
  for (int j = 0; j < K_DIM / 256; j++) {
    Lv[j] = row[threadIdx.x + 256 * j];
    mx = fmaxf(mx, Lv[j]);
  }
  mx = blockMax256(mx);
  const float pm = mx * INV_TP;          // max of pred = max(L)/0.12 (positive scale)

  float se = 0.0f;
#pragma unroll
  for (int j = 0; j < K_DIM / 256; j++) se += __expf(Lv[j] * INV_TP - pm);
  se = blockSum256(se);
  const float lse = pm + __logf(se);     // logsumexp of pred row

  const float rb = rvec[b];
  float dot = 0.0f, ts = 0.0f;
#pragma unroll
  for (int j = 0; j < K_DIM / 256; j++) {
    const int k = threadIdx.x + 256 * j;
    const float t = rb * __expf(Lv[j] * INV_TT + EXP_MAX_C - shift[k]) * cvec[k];
    row[k] = t;                          // assignment overwrites logit in place
    dot += t * (Lv[j] * INV_TP);
    ts += t;
  }
  dot = blockSum256(dot);
  ts = blockSum256(ts);
  if (threadIdx.x == 0)
    lossRows[rowOffset + b] = ts * lse - dot;  // -sum(tgt * logp)
}

__global__ __launch_bounds__(256) void loss_reduce_kernel(
    const float* __restrict__ lossRows, float* __restrict__ outLoss) {
  float s = 0.0f;
  for (int i = threadIdx.x; i < M_TOT; i += 256) s += lossRows[i];
  s = blockSum256(s);
  if (threadIdx.x == 0) outLoss[0] = s * (1.0f / (float)M_TOT);
}

// ---------------- launch ----------------
extern "C" void kernel_launch(void* const* d_in, const int* in_sizes, int n_in,
                              void* d_out, int out_size, void* d_ws, size_t ws_size,
                              hipStream_t stream) {
  (void)in_sizes; (void)n_in; (void)out_size; (void)ws_size;
  const float* x = (const float*)d_in[0];   // [4,4096,768]
  const float* W = (const float*)d_in[1];   // [8192,768]
  float* out = (float*)d_out;               // [4,4096,8192] assignments + [1] loss

  // workspace carve-out (~1.3 MB)
  float* ws = (float*)d_ws;
  float* invn     = ws;                       // 16384
  float* shift    = invn + M_TOT;             // 8192  (per-s, reused)
  float* cvec     = shift + K_DIM;            // 8192  (per-s, reused)
  float* rvec     = cvec + K_DIM;             // 4096  (per-s, reused)
  float* part     = rvec + B_DIM;             // 32*8192
  float* lossRows = part + NCHUNK * K_DIM;    // 16384

  row_invnorm_kernel<<<M_TOT, 256, 0, stream>>>(x, invn);
  gemm_wmma_f32<<<dim3(K_DIM / BN, M_TOT / BM), 256, 0, stream>>>(x, W, invn, out);

  // process one 128 MB s-slice at a time so all sinkhorn passes stay L2-resident
  for (int s = 0; s < S_DIM; s++) {
    float* Ls = out + (size_t)s * B_DIM * K_DIM;
    colmax_part_kernel<<<dim3(K_DIM / 256, NCHUNK), 256, 0, stream>>>(Ls, part);
    colmax_fin_kernel<<<K_DIM / 256, 256, 0, stream>>>(part, shift, cvec);
    init_r_kernel<<<B_DIM / 256, 256, 0, stream>>>(rvec);
    for (int it = 0; it < 3; ++it) {
      colsum_part_kernel<<<dim3(K_DIM / 256, NCHUNK), 256, 0, stream>>>(Ls, rvec, shift, part);
      colsum_fin_kernel<<<K_DIM / 256, 256, 0, stream>>>(part, cvec);
      rowsum_update_kernel<<<B_DIM, 256, 0, stream>>>(Ls, shift, cvec, rvec);
    }
    final_fused_kernel<<<B_DIM, 256, 0, stream>>>(Ls, shift, cvec, rvec, lossRows, s * B_DIM);
  }

  loss_reduce_kernel<<<1, 256, 0, stream>>>(lossRows, out + (size_t)M_TOT * K_DIM);
}